// SparseLinear_12378095747072
// MI455X (gfx1250) — compile-verified
//
#include <hip/hip_runtime.h>

// SparseLinear on MI455X (gfx1250):  y = x @ (S_T + S_N)^T + b
//
// Strategy:
//   1) densify (S_T + S_N)^T into WT[4096][4096] fp32 in d_ws (64 MB -- fits
//      in the 192 MB L2; duplicates sum via global_atomic_add_f32)
//   2) dense fp32 GEMM with V_WMMA_F32_16X16X4_F32 (matches fp32 reference
//      numerics), B-slabs staged to LDS with gfx1250 async global->LDS copies
//      (ASYNCcnt / s_wait_asynccnt), double buffered.
//
// d_in order: x(256x4096 f32), st_idx(2x500000 i32), st_vals(500000 f32),
//             sn_idx(2x100000 i32), sn_vals(100000 f32), b(4096 f32)
// d_ws: needs 64 MB.

#define IN_F  4096
#define OUT_F 4096
#define BATCH 256
#define KC    64          // K-chunk staged to LDS per block
#define NPAD  40          // LDS row stride (floats): 2*40 mod 64 = 16 -> the
                          // two half-wave K-rows hit disjoint bank groups

typedef __attribute__((ext_vector_type(2))) float v2f;
typedef __attribute__((ext_vector_type(8))) float v8f;
typedef __attribute__((ext_vector_type(4))) int   v4i;

#define AS1 __attribute__((address_space(1)))
#define AS3 __attribute__((address_space(3)))

#if defined(__gfx1250__) && __has_builtin(__builtin_amdgcn_global_load_async_to_lds_b128)
#define USE_ASYNC 1
#else
#define USE_ASYNC 0
#endif

// 16-byte global -> LDS copy (async on gfx1250, sync fallback otherwise)
__device__ __forceinline__ void cp16_g2l(const float* g, float* l) {
#if USE_ASYNC
  __builtin_amdgcn_global_load_async_to_lds_b128(
      (AS1 v4i*)const_cast<float*>(g), (AS3 v4i*)l, 0, 0);
#else
  *(float4*)l = *(const float4*)g;
#endif
}

__device__ __forceinline__ void wait_async_le2() {
#if USE_ASYNC
  asm volatile("s_wait_asynccnt 0x2" ::: "memory");
#endif
}

__device__ __forceinline__ void wait_async_0() {
#if USE_ASYNC
  asm volatile("s_wait_asynccnt 0x0" ::: "memory");
#endif
}

// ---------------------------------------------------------------- zero WT ---
__global__ void __launch_bounds__(256) zero_f4(float4* __restrict__ p, int n4) {
  int i = blockIdx.x * blockDim.x + threadIdx.x;
  if (i < n4) p[i] = make_float4(0.f, 0.f, 0.f, 0.f);
}

// --------------------------------------------------- scatter COO -> dense ---
// idx flattened (2, nnz): rows = idx[0..nnz), cols = idx[nnz..2nnz).
__global__ void __launch_bounds__(256) scatter_coo(const int* __restrict__ idx,
                                                   const float* __restrict__ vals,
                                                   int nnz,
                                                   float* __restrict__ WT) {
  int i = blockIdx.x * blockDim.x + threadIdx.x;
  if (i >= nnz) return;
  int r = idx[i];        // output-feature row of W
  int c = idx[nnz + i];  // input-feature col of W
  unsafeAtomicAdd(&WT[(size_t)c * OUT_F + r], vals[i]);
}

// ------------------------------------------------------- fp32 WMMA GEMM -----
// Block = 256 threads (8 wave32). Block owns 32 N-columns; wave w owns M rows
// [32w, 32w+32). Per K-chunk (KC=64) the block stages WT[k0..k0+63][n0..n0+31]
// into LDS (async, double buffered); every wave then reads shared B fragments
// from LDS and runs 2x2 tiles of V_WMMA_F32_16X16X4_F32 over the chunk.
//
// A (16x4 fp32) lane layout: half = lane>>4, r = lane&15
//   {VGPR0,VGPR1} = x[m0+r][k+2*half .. k+2*half+1]  -> one float2 load
// B (4x16 fp32) lane layout (symmetric):
//   {VGPR0,VGPR1} = WT[k+2*half .. +1][n0+r]         -> two ds_load_b32
// C/D (16x16 fp32): VGPR i -> row (i + 8*half), col r of the tile.
__global__ void __launch_bounds__(256) spmm_wmma(const float* __restrict__ x,
                                                 const float* __restrict__ WT,
                                                 const float* __restrict__ bias,
                                                 float* __restrict__ out) {
  __shared__ float slab[2][KC][NPAD];

  const int tid  = threadIdx.x;
  const int lane = tid & 31;
  const int wave = tid >> 5;            // 0..7 -> M block of 32 rows
  const int half = lane >> 4;
  const int r    = lane & 15;
  const int n0   = blockIdx.x * 32;     // 128 blocks cover N = 4096
  const int m0   = wave * 32;           // 8 waves cover M = 256

  // Slab loader: 64 rows x 32 floats = 512 float4 slots, 2 per thread.
  const int row0 = tid >> 3;            // 0..31
  const int cc   = (tid & 7) * 4;       // 0,4,...,28
  const int row1 = row0 + 32;           // 32..63
  const float* gs0 = WT + (size_t)row0 * OUT_F + n0 + cc;
  const float* gs1 = WT + (size_t)row1 * OUT_F + n0 + cc;

  v8f c00 = {}; v8f c01 = {}; v8f c10 = {}; v8f c11 = {};

  const float* xr0 = x + (size_t)(m0 + r) * IN_F;        // A tile 0 rows
  const float* xr1 = x + (size_t)(m0 + 16 + r) * IN_F;   // A tile 1 rows

  const int NCHUNK = IN_F / KC;         // 64

  // Prologue: stage chunk 0 into buffer 0.
  cp16_g2l(gs0, &slab[0][row0][cc]);
  cp16_g2l(gs1, &slab[0][row1][cc]);

  for (int ck = 0; ck < NCHUNK; ++ck) {
    const int cur = ck & 1;
    const int nk  = ck + 1;
    if (nk < NCHUNK) {
      const size_t gofs = (size_t)nk * KC * OUT_F;
      cp16_g2l(gs0 + gofs, &slab[nk & 1][row0][cc]);
      cp16_g2l(gs1 + gofs, &slab[nk & 1][row1][cc]);
      wait_async_le2();                 // oldest 2 (current chunk) landed
    } else {
      wait_async_0();
    }
    __syncthreads();                    // slab[cur] visible to all waves

    const float* xk0 = xr0 + ck * KC + 2 * half;
    const float* xk1 = xr1 + ck * KC + 2 * half;
    const float* S0  = &slab[cur][0][r];        // + row*NPAD
    const float* S1  = &slab[cur][0][16 + r];

#pragma unroll 4
    for (int kk = 0; kk < KC; kk += 4) {
      const int ka = kk + 2 * half;

      v2f a0 = *(const v2f*)(xk0 + kk);         // {K=2h, K=2h+1}
      v2f a1 = *(const v2f*)(xk1 + kk);

      v2f b0, b1;
      b0.x = S0[ka * NPAD];
      b0.y = S0[(ka + 1) * NPAD];
      b1.x = S1[ka * NPAD];
      b1.y = S1[(ka + 1) * NPAD];

      c00 = __builtin_amdgcn_wmma_f32_16x16x4_f32(false, a0, false, b0,
                                                  (short)0, c00, false, false);
      c01 = __builtin_amdgcn_wmma_f32_16x16x4_f32(false, a0, false, b1,
                                                  (short)0, c01, false, false);
      c10 = __builtin_amdgcn_wmma_f32_16x16x4_f32(false, a1, false, b0,
                                                  (short)0, c10, false, false);
      c11 = __builtin_amdgcn_wmma_f32_16x16x4_f32(false, a1, false, b1,
                                                  (short)0, c11, false, false);
    }
    __syncthreads();                    // all waves done reading slab[cur]
  }

  // Store 32x32 block with fused bias. C/D layout: VGPR i -> row i + 8*half.
  const float bn0 = bias[n0 + r];
  const float bn1 = bias[n0 + 16 + r];
  const int mr0 = m0 + 8 * half;
  const int mr1 = m0 + 16 + 8 * half;
#pragma unroll
  for (int i = 0; i < 8; ++i) {
    out[(size_t)(mr0 + i) * OUT_F + n0 + r]      = c00[i] + bn0;
    out[(size_t)(mr0 + i) * OUT_F + n0 + 16 + r] = c01[i] + bn1;
    out[(size_t)(mr1 + i) * OUT_F + n0 + r]      = c10[i] + bn0;
    out[(size_t)(mr1 + i) * OUT_F + n0 + 16 + r] = c11[i] + bn1;
  }
}

// ----------------------------------------------------------------- launch ---
extern "C" void kernel_launch(void* const* d_in, const int* in_sizes, int n_in,
                              void* d_out, int out_size, void* d_ws, size_t ws_size,
                              hipStream_t stream) {
  const float* x       = (const float*)d_in[0];
  const int*   st_idx  = (const int*)d_in[1];
  const float* st_vals = (const float*)d_in[2];
  const int*   sn_idx  = (const int*)d_in[3];
  const float* sn_vals = (const float*)d_in[4];
  const float* bias    = (const float*)d_in[5];
  float*       out     = (float*)d_out;
  float*       WT      = (float*)d_ws;   // 4096*4096*4 = 64 MB scratch

  const int nnz_t = in_sizes[2];
  const int nnz_n = in_sizes[4];

  // 1) WT = 0
  const int n4 = (IN_F * OUT_F) / 4;
  zero_f4<<<(n4 + 255) / 256, 256, 0, stream>>>((float4*)WT, n4);

  // 2) scatter both sparse matrices (duplicates sum)
  scatter_coo<<<(nnz_t + 255) / 256, 256, 0, stream>>>(st_idx, st_vals, nnz_t, WT);
  scatter_coo<<<(nnz_n + 255) / 256, 256, 0, stream>>>(sn_idx, sn_vals, nnz_n, WT);

  // 3) dense fp32 WMMA GEMM + bias: out = x @ WT + b
  spmm_wmma<<<OUT_F / 32, 256, 0, stream>>>(x, WT, bias, out);
}